// Polychoron600Quantizer_88304527606120
// MI455X (gfx1250) — compile-verified
//
#include <hip/hip_runtime.h>

// Polychoron (600-cell) nearest-vertex quantizer for MI455X / gfx1250.
//
// All 120 vertices of the 600-cell lie on the unit 3-sphere, so
//   argmin_m ||x - v_m||^2  ==  argmax_m  x . v_m
// evaluated as a tiny GEMM with V_WMMA_F32_16X16X4_F32 (K=4 exactly).
//
// A operand = 16 vertices x 4 coords (8 tiles cover 128 rows, zero padded)
// B operand = 4 coords x 16 points
// D[m][n]   = v_m . x_n : vertices stripe across VGPRs, points across lanes.
// Per-point argmax: pairwise TREE over the 8 D registers (depth 3) and then
// as a sequential chain over the 8 tiles (7 steps). Each tree node is v_cmp + two v_cndmask that
// share VCC and dual-issue (no fmaxf -> no canonicalize ops). Indices are
// tracked as the compile-time-constant relative slot (t*16+r, inline-constant
// cndmask operands) and the runtime half*8 offset is added once at the end,
// so no index-constant VGPRs get hoisted.

typedef __attribute__((ext_vector_type(2))) float v2f;
typedef __attribute__((ext_vector_type(4))) float v4f;
typedef __attribute__((ext_vector_type(8))) float v8f;

#ifndef NVERT
#define NVERT 120
#endif
#define NVPAD 128  // 8 WMMA tiles of 16 vertex rows

// Combine two (score, idx) candidates; keep the lower index on exact ties.
// One v_cmp + two v_cndmask sharing the compare result.
__device__ __forceinline__ void comb(float sa, int ra, float sb, int rb,
                                     float& so, int& ro) {
    const bool gt = sb > sa;
    ro = gt ? rb : ra;
    so = gt ? sb : sa;
}

// Tree-argmax over one 8-slot WMMA D tile; rel indices are t*16 + r.
__device__ __forceinline__ void tile_argmax(const v8f& d, int t, float& ts,
                                            int& tr) {
    float s01, s23, s45, s67, s03, s47;
    int r01, r23, r45, r67, r03, r47;
    const int b = t * 16;
    comb(d[0], b + 0, d[1], b + 1, s01, r01);
    comb(d[2], b + 2, d[3], b + 3, s23, r23);
    comb(d[4], b + 4, d[5], b + 5, s45, r45);
    comb(d[6], b + 6, d[7], b + 7, s67, r67);
    comb(s01, r01, s23, r23, s03, r03);
    comb(s45, r45, s67, r67, s47, r47);
    comb(s03, r03, s47, r47, ts, tr);
}

__global__ __launch_bounds__(256) void poly600_quantize_kernel(
    const float* __restrict__ x,      // [npoints, 4]
    const float* __restrict__ verts,  // [120, 4]
    float* __restrict__ out,          // [npoints, 4]
    int npoints)
{
    // Zero-padded vertex table in LDS (2 KB): feeds the A-operand register
    // loads and the final winner gather.
    __shared__ v4f lv[NVPAD];
    const int tid = threadIdx.x;
    if (tid < NVPAD) {
        v4f v = {0.f, 0.f, 0.f, 0.f};
        if (tid < NVERT) v = *(const v4f*)(verts + tid * 4);
        lv[tid] = v;
    }
    __syncthreads();

    const int lane = tid & 31;
    const int col  = lane & 15;  // point within tile / vertex row within tile
    const int half = lane >> 4;  // 0: K=0..1 (M rows 0..7); 1: K=2..3 (M rows 8..15)

    // A tiles: lane L<16 holds (v[m][0], v[m][1]) for m = t*16+L,
    //          lane L>=16 holds (v[m][2], v[m][3]) for m = t*16+(L-16).
    v2f a[8];
#pragma unroll
    for (int t = 0; t < 8; ++t) {
        const v2f* vp = (const v2f*)&lv[t * 16 + col];
        a[t] = vp[half];
    }

    const int wave   = blockIdx.x * (blockDim.x >> 5) + (tid >> 5);
    const int nwaves = gridDim.x * (blockDim.x >> 5);
    const v8f czero  = {0.f, 0.f, 0.f, 0.f, 0.f, 0.f, 0.f, 0.f};
    const int mbase  = half * 8;  // runtime index offset, added once at the end
    const long long stride = (long long)nwaves * 32;

    // Each wave iteration: two 16-point tiles -> 16 WMMAs, all 32 lanes store.
    for (long long base = (long long)wave * 32; base + 31 < (long long)npoints;
         base += stride) {
        // Prefetch next iteration's points (global_prefetch_b8).
        __builtin_prefetch(x + (base + stride) * 4, 0, 1);

        // B operand: 8B per lane, fully coalesced over the wave.
        v2f b0 = *(const v2f*)(x + (base + col) * 4 + half * 2);
        v2f b1 = *(const v2f*)(x + (base + 16 + col) * 4 + half * 2);

        float s0, s1;       // running best score per point
        int   rel0, rel1;   // running best relative slot (t*16 + r)
#pragma unroll
        for (int t = 0; t < 8; ++t) {
            v8f d0 = __builtin_amdgcn_wmma_f32_16x16x4_f32(
                false, a[t], false, b0, (short)0, czero, false, false);
            v8f d1 = __builtin_amdgcn_wmma_f32_16x16x4_f32(
                false, a[t], false, b1, (short)0, czero, false, false);
            float ts0, ts1;
            int   tr0, tr1;
            tile_argmax(d0, t, ts0, tr0);
            tile_argmax(d1, t, ts1, tr1);
            if (t == 0) {
                s0 = ts0; rel0 = tr0;
                s1 = ts1; rel1 = tr1;
            } else {
                comb(s0, rel0, ts0, tr0, s0, rel0);
                comb(s1, rel1, ts1, tr1, s1, rel1);
            }
        }

        // Materialize the absolute vertex index (single add per point-tile).
        int i0 = rel0 + mbase;
        int i1 = rel1 + mbase;

        // Merge the two lane halves (vertex rows 0..7 vs 8..15 of each tile);
        // prefer the lower index on exact ties to match argmin semantics.
        float so0 = __shfl_xor(s0, 16, 32);
        int   io0 = __shfl_xor(i0, 16, 32);
        if (so0 > s0 || (so0 == s0 && io0 < i0)) { i0 = io0; }
        float so1 = __shfl_xor(s1, 16, 32);
        int   io1 = __shfl_xor(i1, 16, 32);
        if (so1 > s1 || (so1 == s1 && io1 < i1)) { i1 = io1; }

        // Lanes 0..15 own tile-0 results, lanes 16..31 own tile-1 results:
        // every lane stores one 16B winner -> 512B contiguous per wave.
        const int idx = (lane < 16) ? i0 : i1;
        *(v4f*)(out + (base + lane) * 4) = lv[idx];
    }
}

extern "C" void kernel_launch(void* const* d_in, const int* in_sizes, int n_in,
                              void* d_out, int out_size, void* d_ws, size_t ws_size,
                              hipStream_t stream) {
    (void)n_in; (void)d_ws; (void)ws_size; (void)out_size;
    const float* x     = (const float*)d_in[0];  // [512*4096*4] f32
    const float* verts = (const float*)d_in[1];  // [120*4]      f32
    float*       out   = (float*)d_out;          // same shape as x

    const int npoints = in_sizes[0] / 4;         // 2,097,152

    // 256 threads = 8 waves/block; each wave covers 32 points/iteration.
    const int block = 256;
    long long needed = ((long long)npoints + 255) / 256;
    int grid = (int)(needed < 2048 ? (needed > 0 ? needed : 1) : 2048);

    poly600_quantize_kernel<<<grid, block, 0, stream>>>(x, verts, out, npoints);
}